// Set2set_71743133712505
// MI455X (gfx1250) — compile-verified
//
#include <hip/hip_runtime.h>

typedef __attribute__((ext_vector_type(2))) float v2f;
typedef __attribute__((ext_vector_type(4))) float vf4;
typedef __attribute__((ext_vector_type(8))) float v8f;

#define NN    200000
#define CC    256
#define BB    512
#define NSTEP 3

__device__ __forceinline__ float sigmoidf_(float x) { return 1.0f / (1.0f + __expf(-x)); }

// ---------------------------------------------------------------- init
__global__ void k_init(float* __restrict__ qs0, float* __restrict__ c_state) {
    int i = blockIdx.x * blockDim.x + threadIdx.x;
    if (i < BB * 2 * CC) qs0[i] = 0.0f;
    if (i < BB * CC)     c_state[i] = 0.0f;
}

// starts[b] = first n with batch[n] >= b  (batch sorted); starts[BB] = NN
__global__ void k_starts(const long long* __restrict__ batch, int* __restrict__ starts) {
    int b = blockIdx.x * blockDim.x + threadIdx.x;
    if (b > BB) return;
    int lo = 0, hi = NN;
    while (lo < hi) {
        int mid = (lo + hi) >> 1;
        if ((int)batch[mid] < b) lo = mid + 1; else hi = mid;
    }
    starts[b] = lo;
}

// ---------------------------------------------------------------- LSTM (WMMA f32 16x16x4)
// gates[512,1024] = qin[512,512] @ W_ih^T + qin[:, :256] @ W_hh^T + (b_ih+b_hh)
// One wave per (row-tile tm, col-tile-in-C tc); 4 accumulators = i,f,g,o tiles.
__global__ void __launch_bounds__(32) k_lstm(
    const float* __restrict__ qin, const float* __restrict__ W_ih,
    const float* __restrict__ W_hh, const float* __restrict__ b_ih,
    const float* __restrict__ b_hh, float* __restrict__ c_state,
    float* __restrict__ qout)
{
    const int wid  = blockIdx.x;      // 0..511
    const int tm   = wid >> 4;        // 0..31 : 16-row tile of B=512
    const int tc   = wid & 15;        // 0..15 : 16-col tile of C=256
    const int lane = threadIdx.x;
    const int half = lane >> 4;       // 0: K=k0,k0+1   1: K=k0+2,k0+3
    const int lr   = lane & 15;

    v8f acc0 = {}, acc1 = {}, acc2 = {}, acc3 = {};

    const float* arow = qin + (size_t)(tm * 16 + lr) * (2 * CC);
    const float* w0 = W_ih + (size_t)(0 * CC + tc * 16 + lr) * (2 * CC);
    const float* w1 = W_ih + (size_t)(1 * CC + tc * 16 + lr) * (2 * CC);
    const float* w2 = W_ih + (size_t)(2 * CC + tc * 16 + lr) * (2 * CC);
    const float* w3 = W_ih + (size_t)(3 * CC + tc * 16 + lr) * (2 * CC);

    // q_star @ W_ih^T   (K = 2C = 512)
    for (int k0 = 0; k0 < 2 * CC; k0 += 4) {
        const int ko = k0 + 2 * half;
        v2f a = *(const v2f*)(arow + ko);
        v2f b;
        b = *(const v2f*)(w0 + ko);
        acc0 = __builtin_amdgcn_wmma_f32_16x16x4_f32(false, a, false, b, (short)0, acc0, false, false);
        b = *(const v2f*)(w1 + ko);
        acc1 = __builtin_amdgcn_wmma_f32_16x16x4_f32(false, a, false, b, (short)0, acc1, false, false);
        b = *(const v2f*)(w2 + ko);
        acc2 = __builtin_amdgcn_wmma_f32_16x16x4_f32(false, a, false, b, (short)0, acc2, false, false);
        b = *(const v2f*)(w3 + ko);
        acc3 = __builtin_amdgcn_wmma_f32_16x16x4_f32(false, a, false, b, (short)0, acc3, false, false);
    }

    // h @ W_hh^T   (h == qin[:, :C], K = C = 256)
    const float* v0 = W_hh + (size_t)(0 * CC + tc * 16 + lr) * CC;
    const float* v1 = W_hh + (size_t)(1 * CC + tc * 16 + lr) * CC;
    const float* v2 = W_hh + (size_t)(2 * CC + tc * 16 + lr) * CC;
    const float* v3 = W_hh + (size_t)(3 * CC + tc * 16 + lr) * CC;
    for (int k0 = 0; k0 < CC; k0 += 4) {
        const int ko = k0 + 2 * half;
        v2f a = *(const v2f*)(arow + ko);
        v2f b;
        b = *(const v2f*)(v0 + ko);
        acc0 = __builtin_amdgcn_wmma_f32_16x16x4_f32(false, a, false, b, (short)0, acc0, false, false);
        b = *(const v2f*)(v1 + ko);
        acc1 = __builtin_amdgcn_wmma_f32_16x16x4_f32(false, a, false, b, (short)0, acc1, false, false);
        b = *(const v2f*)(v2 + ko);
        acc2 = __builtin_amdgcn_wmma_f32_16x16x4_f32(false, a, false, b, (short)0, acc2, false, false);
        b = *(const v2f*)(v3 + ko);
        acc3 = __builtin_amdgcn_wmma_f32_16x16x4_f32(false, a, false, b, (short)0, acc3, false, false);
    }

    // epilogue: bias + gate nonlinearity + cell update, fused in-register
    const int col = tc * 16 + lr;                     // C/D layout: N = lane&15
    const float bi0 = b_ih[0 * CC + col] + b_hh[0 * CC + col];
    const float bi1 = b_ih[1 * CC + col] + b_hh[1 * CC + col];
    const float bi2 = b_ih[2 * CC + col] + b_hh[2 * CC + col];
    const float bi3 = b_ih[3 * CC + col] + b_hh[3 * CC + col];

#pragma unroll
    for (int r = 0; r < 8; ++r) {
        const int m = tm * 16 + r + 8 * half;         // C/D layout: M = vgpr + 8*(lane>=16)
        float gi = sigmoidf_(acc0[r] + bi0);
        float gf = sigmoidf_(acc1[r] + bi1);
        float gg = tanhf(acc2[r] + bi2);
        float go = sigmoidf_(acc3[r] + bi3);
        float cp = c_state[(size_t)m * CC + col];
        float cn = gf * cp + gi * gg;
        c_state[(size_t)m * CC + col] = cn;
        qout[(size_t)m * (2 * CC) + col] = go * tanhf(cn);   // q half of q_star
    }
}

// ---------------------------------------------------------------- fused attention
// One block per graph (segment is contiguous: batch sorted). 8 waves split the
// segment's nodes; each wave runs an online (streaming) softmax keeping
// (m, d, racc[8 floats/lane]) in registers — single pass over x. Wave partials
// merge in LDS with log-sum-exp rescaling. r written into qs_out[:, C:2C].
__global__ void __launch_bounds__(256) k_attend(
    const float* __restrict__ x, const int* __restrict__ starts,
    float* __restrict__ qs_out)
{
    const int b    = blockIdx.x;
    const int tid  = threadIdx.x;
    const int w    = tid >> 5;          // wave 0..7
    const int lane = tid & 31;

    const int s = starts[b];
    const int t = starts[b + 1];

    // q row for this graph: lane covers cols [lane*4..+3] and [128+lane*4..+3]
    const vf4* qr = (const vf4*)(qs_out + (size_t)b * (2 * CC));
    const vf4 qa = qr[lane];
    const vf4 qb = qr[lane + 32];

    float m = -__builtin_inff();
    float d = 0.0f;
    vf4 r0 = {}, r1 = {};

    // wave w owns node pairs (s+2w+16k, s+2w+16k+1); 2-node unroll for ILP
    int n = s + 2 * w;
    for (; n + 1 < t; n += 16) {
        const vf4* xr0 = (const vf4*)(x + (size_t)n * CC);
        const vf4* xr1 = (const vf4*)(x + (size_t)(n + 1) * CC);
        vf4 xa1 = xr0[lane], xb1 = xr0[lane + 32];
        vf4 xa2 = xr1[lane], xb2 = xr1[lane + 32];

        vf4 p1 = xa1 * qa + xb1 * qb;
        vf4 p2 = xa2 * qa + xb2 * qb;
        float e1 = p1.x + p1.y + p1.z + p1.w;
        float e2 = p2.x + p2.y + p2.z + p2.w;
#pragma unroll
        for (int off = 16; off; off >>= 1) {
            e1 += __shfl_xor(e1, off, 32);
            e2 += __shfl_xor(e2, off, 32);
        }
        float mn  = fmaxf(m, fmaxf(e1, e2));
        float sc  = __expf(m - mn);          // m=-inf first iter -> 0
        float ex1 = __expf(e1 - mn);
        float ex2 = __expf(e2 - mn);
        d  = d * sc + ex1 + ex2;
        r0 = r0 * sc + ex1 * xa1 + ex2 * xa2;
        r1 = r1 * sc + ex1 * xb1 + ex2 * xb2;
        m = mn;
    }
    if (n < t) {                              // odd tail node (at most one wave)
        const vf4* xr0 = (const vf4*)(x + (size_t)n * CC);
        vf4 xa1 = xr0[lane], xb1 = xr0[lane + 32];
        vf4 p1 = xa1 * qa + xb1 * qb;
        float e1 = p1.x + p1.y + p1.z + p1.w;
#pragma unroll
        for (int off = 16; off; off >>= 1) e1 += __shfl_xor(e1, off, 32);
        float mn = fmaxf(m, e1);
        float sc = __expf(m - mn);
        float ex = __expf(e1 - mn);
        d  = d * sc + ex;
        r0 = r0 * sc + ex * xa1;
        r1 = r1 * sc + ex * xb1;
        m = mn;
    }

    // -------- merge 8 wave partials in LDS --------
    __shared__ float sm[8], sd[8], sr[CC];
    sr[tid] = 0.0f;                           // blockDim == CC
    if (lane == 0) { sm[w] = m; sd[w] = d; }
    __syncthreads();

    float M = sm[0];
#pragma unroll
    for (int i = 1; i < 8; ++i) M = fmaxf(M, sm[i]);

    const float scw = (d > 0.0f) ? __expf(m - M) : 0.0f;   // guard empty wave (NaN from -inf - -inf)
#pragma unroll
    for (int j = 0; j < 4; ++j) {
        atomicAdd(&sr[lane * 4 + j],      r0[j] * scw);    // ds_add_f32
        atomicAdd(&sr[128 + lane * 4 + j], r1[j] * scw);
    }
    __syncthreads();

    float D = 0.0f;
#pragma unroll
    for (int i = 0; i < 8; ++i)
        if (sd[i] > 0.0f) D += sd[i] * __expf(sm[i] - M);

    // empty segment -> D == 0 -> r = 0 (matches segment_sum over empty set)
    qs_out[(size_t)b * (2 * CC) + CC + tid] = (D > 0.0f) ? sr[tid] / D : 0.0f;
}

// ---------------------------------------------------------------- driver
extern "C" void kernel_launch(void* const* d_in, const int* in_sizes, int n_in,
                              void* d_out, int out_size, void* d_ws, size_t ws_size,
                              hipStream_t stream)
{
    const float*     x     = (const float*)d_in[0];
    const long long* batch = (const long long*)d_in[1];
    const float*     W_ih  = (const float*)d_in[2];
    const float*     W_hh  = (const float*)d_in[3];
    const float*     b_ih  = (const float*)d_in[4];
    const float*     b_hh  = (const float*)d_in[5];

    float* qs0     = (float*)d_ws;                 // [512][512]
    float* qs1     = qs0 + BB * 2 * CC;            // [512][512]
    float* c_state = qs1 + BB * 2 * CC;            // [512][256]
    int*   starts  = (int*)(c_state + BB * CC);    // [513]

    k_init<<<(BB * 2 * CC + 255) / 256, 256, 0, stream>>>(qs0, c_state);
    k_starts<<<(BB + 1 + 127) / 128, 128, 0, stream>>>(batch, starts);

    for (int s = 0; s < NSTEP; ++s) {
        float* qin  = (s & 1) ? qs1 : qs0;
        float* qout = (s & 1) ? qs0 : qs1;
        k_lstm<<<BB, 32, 0, stream>>>(qin, W_ih, W_hh, b_ih, b_hh, c_state, qout);
        k_attend<<<BB, CC, 0, stream>>>(x, starts, qout);
    }
    // after steps 0,1,2 the live buffer is qs1
    hipMemcpyAsync(d_out, qs1, (size_t)BB * 2 * CC * sizeof(float),
                   hipMemcpyDeviceToDevice, stream);
}